// Coach_20538533609986
// MI455X (gfx1250) — compile-verified
//
#include <hip/hip_runtime.h>

typedef __attribute__((ext_vector_type(2))) float v2f;
typedef __attribute__((ext_vector_type(8))) float v8f;

// ---------------- degree / norm ----------------

__global__ void init_deg_kernel(float* __restrict__ deg, int n) {
    int i = blockIdx.x * blockDim.x + threadIdx.x;
    if (i < n) deg[i] = 1.0f;  // self-loop contribution
}

__global__ void edge_deg_kernel(const int* __restrict__ dst, float* __restrict__ deg, int E) {
    int e = blockIdx.x * blockDim.x + threadIdx.x;
    if (e < E) atomicAdd(&deg[dst[e]], 1.0f);
}

__global__ void dinv_kernel(float* __restrict__ deg, int n) {
    int i = blockIdx.x * blockDim.x + threadIdx.x;
    if (i < n) deg[i] = rsqrtf(deg[i]);  // deg >= 1 always (self-loops)
}

// ---------------- WMMA f32 GEMM: H[N,FOUT] = X[N,FIN] @ W[FIN,FOUT] ----------------
// One wave computes a 16-row tile across all FOUT columns using V_WMMA_F32_16X16X4_F32.
// A frag (16x4 f32, 2 VGPRs): lanes 0-15 -> K=0,1 ; lanes 16-31 -> K=2,3 ; M = lane%16.
// B frag (4x16 f32, 2 VGPRs): lanes 0-15 -> K=0,1 ; lanes 16-31 -> K=2,3 ; N = lane%16.
// C/D (16x16 f32, 8 VGPRs): VGPR v -> row v + 8*(lane>=16), col = lane%16.
//
// W is staged in LDS, pre-swizzled as K-pair-interleaved v2f so that each B fragment
// is one aligned ds_load_b64 landing directly in an adjacent VGPR pair (no repack movs).
// Pair-row stride padded +16 pairs (128 B) to keep the two half-wave rows on disjoint banks.

template <int FIN, int FOUT>
__global__ __launch_bounds__(128) void gemm_wmma_kernel(const float* __restrict__ X,
                                                        const float* __restrict__ W,
                                                        float* __restrict__ H, int n) {
    constexpr int CT   = FOUT / 16;   // column tiles
    constexpr int KP   = FIN / 2;     // number of K-pairs
    constexpr int PSTR = FOUT + 16;   // padded pair-row stride (v2f units)

    __shared__ v2f sW2[KP * PSTR];
    for (int i = threadIdx.x; i < KP * FOUT; i += 128) {
        const int kk  = i / FOUT;         // K-pair index (FOUT is a power of two -> shift)
        const int col = i & (FOUT - 1);
        v2f w;
        w.x = W[(2 * kk) * FOUT + col];
        w.y = W[(2 * kk + 1) * FOUT + col];
        sW2[kk * PSTR + col] = w;
    }
    __syncthreads();

    const int lane = threadIdx.x & 31;
    const int wave = __builtin_amdgcn_readfirstlane(threadIdx.x >> 5);  // SGPR -> scalar branches
    const int tile = blockIdx.x * 4 + wave;          // 16-row tile index (wave-uniform, scalar)
    if (tile * 16 >= n) return;                      // scalar branch: EXEC stays full
    const int nlo = lane & 15;
    const int kh  = (lane >> 4) << 1;                // 0 or 2
    int row = tile * 16 + nlo;
    if (row >= n) row = n - 1;                       // clamp loads for partial tiles

    // Speculative prefetch of this row of X (gfx1250 global_prefetch_b8).
    __builtin_prefetch(X + (size_t)row * FIN, 0, 3);

    v8f acc[CT] = {};

    for (int k = 0; k < FIN; k += 4) {
        v2f a;
        a.x = X[row * FIN + k + kh];                 // merged to global_load_b64
        a.y = X[row * FIN + k + kh + 1];
        const v2f* wrow = &sW2[((k + kh) >> 1) * PSTR + nlo];
#pragma unroll
        for (int c = 0; c < CT; ++c) {
            v2f b = wrow[c * 16];                    // one ds_load_b64 per fragment
            acc[c] = __builtin_amdgcn_wmma_f32_16x16x4_f32(
                false, a, false, b, (short)0, acc[c], false, false);
        }
    }

    const int rbase = tile * 16 + ((lane >> 4) << 3);
    if (tile * 16 + 16 <= n) {
        // Full tile (the common case: N % 16 == 0) -> branch-free stores, scalar guard.
        float* __restrict__ Hp = H + (size_t)rbase * FOUT + nlo;
#pragma unroll
        for (int c = 0; c < CT; ++c)
#pragma unroll
            for (int v = 0; v < 8; ++v)
                Hp[(size_t)v * FOUT + c * 16] = acc[c][v];
    } else {
        // Cold path: partial last tile.
#pragma unroll
        for (int c = 0; c < CT; ++c)
#pragma unroll
            for (int v = 0; v < 8; ++v)
                if (rbase + v < n) H[(size_t)(rbase + v) * FOUT + c * 16 + nlo] = acc[c][v];
    }
}

// ---------------- aggregation ----------------

// agg[n,f] = H[n,f] * dinv[n]^2   (self-loop message; also initializes agg)
template <int F>
__global__ void self_msg_kernel(const float* __restrict__ H, const float* __restrict__ dinv,
                                float* __restrict__ agg, int n) {
    int g = blockIdx.x * blockDim.x + threadIdx.x;
    if (g >= n * F) return;
    int node = g / F;
    float d = dinv[node];
    agg[g] = H[g] * d * d;
}

// agg[dst,f] += H[src,f] * dinv[src]*dinv[dst]
template <int F>
__global__ void edge_scatter_kernel(const int* __restrict__ src, const int* __restrict__ dst,
                                    const float* __restrict__ dinv, const float* __restrict__ H,
                                    float* __restrict__ agg, int E) {
    int g = blockIdx.x * blockDim.x + threadIdx.x;
    if (g >= E * F) return;
    int e = g / F;          // F is a power of two -> shifts
    int f = g & (F - 1);
    int s = src[e];
    int d = dst[e];
    float w = dinv[s] * dinv[d];
    atomicAdd(&agg[d * F + f], H[s * F + f] * w);
}

template <int F>
__global__ void bias_relu_kernel(float* __restrict__ A, const float* __restrict__ b, int n) {
    int g = blockIdx.x * blockDim.x + threadIdx.x;
    if (g >= n * F) return;
    float v = A[g] + b[g & (F - 1)];
    A[g] = v > 0.0f ? v : 0.0f;
}

// ---------------- launcher ----------------

extern "C" void kernel_launch(void* const* d_in, const int* in_sizes, int n_in,
                              void* d_out, int out_size, void* d_ws, size_t ws_size,
                              hipStream_t stream) {
    const float* X  = (const float*)d_in[0];   // [N,128]
    const int*   EI = (const int*)d_in[1];     // [2,E]
    const float* W1 = (const float*)d_in[2];   // [128,64]
    const float* b1 = (const float*)d_in[3];   // [64]
    const float* W2 = (const float*)d_in[4];   // [64,32]
    const float* b2 = (const float*)d_in[5];   // [32]
    float* out = (float*)d_out;                // [N,32]

    const int N = in_sizes[0] / 128;
    const int E = in_sizes[1] / 2;
    const int* src = EI;
    const int* dst = EI + E;

    float* ws   = (float*)d_ws;
    float* dinv = ws;                                    // N floats (deg -> dinv in place)
    size_t off  = ((size_t)N + 63) & ~(size_t)63;
    float* bufA = ws + off;                              // N*64 (h1, later h2 uses N*32)
    float* bufB = bufA + (size_t)N * 64;                 // N*64 (agg1 -> x1)

    const int T = 256;
    const int tiles = (N + 15) / 16;
    const int gblocks = (tiles + 3) / 4;

    // normalization
    init_deg_kernel<<<(N + T - 1) / T, T, 0, stream>>>(dinv, N);
    edge_deg_kernel<<<(E + T - 1) / T, T, 0, stream>>>(dst, dinv, E);
    dinv_kernel<<<(N + T - 1) / T, T, 0, stream>>>(dinv, N);

    // layer 1: h1 = X @ W1 ; agg ; +b1 ; relu
    gemm_wmma_kernel<128, 64><<<gblocks, 128, 0, stream>>>(X, W1, bufA, N);
    self_msg_kernel<64><<<(N * 64 + T - 1) / T, T, 0, stream>>>(bufA, dinv, bufB, N);
    edge_scatter_kernel<64><<<(E * 64 + T - 1) / T, T, 0, stream>>>(src, dst, dinv, bufA, bufB, E);
    bias_relu_kernel<64><<<(N * 64 + T - 1) / T, T, 0, stream>>>(bufB, b1, N);

    // layer 2: h2 = x1 @ W2 ; agg into d_out ; +b2 ; relu
    gemm_wmma_kernel<64, 32><<<gblocks, 128, 0, stream>>>(bufB, W2, bufA, N);
    self_msg_kernel<32><<<(N * 32 + T - 1) / T, T, 0, stream>>>(bufA, dinv, out, N);
    edge_scatter_kernel<32><<<(E * 32 + T - 1) / T, T, 0, stream>>>(src, dst, dinv, bufA, out, E);
    bias_relu_kernel<32><<<(N * 32 + T - 1) / T, T, 0, stream>>>(out, b2, N);
}